// HalfGaussianConditionalLogit_56934086476379
// MI455X (gfx1250) — compile-verified
//
#include <hip/hip_runtime.h>
#include <math.h>

// ---------------------------------------------------------------------------
// HalfGaussianConditionalLogit on MI455X (gfx1250, wave32)
//
// Phase 1: norms + 1/norms (wave32 shuffle reduction)
// Phase 2: min over x-norms (single block)
// Phase 3: cos matrix 512x512 (K=64) via V_WMMA_F32_16X16X4_F32, f32 precision
// Phase 4: ln_x column integral (512 threads)
// Phase 5: main 262144-pair integral kernel (transcendental bound)
// ---------------------------------------------------------------------------

#define NX     512
#define NY     512
#define KDIM   64
#define NSAMP  256
#define MAXV   10.0f
#define INFC   1.0e10f
#define LNS2PI 0.9189385332046727f
#define LNDELTA (-3.2386784521643803f)   // log(10/255)

typedef __attribute__((ext_vector_type(2))) float v2f;
typedef __attribute__((ext_vector_type(8))) float v8f;

// ---------------------------------------------------------------------------
// math helpers
// ---------------------------------------------------------------------------
__device__ __forceinline__ float lnpdf(float t) {
    return fmaf(-0.5f * t, t, -LNS2PI);
}

// log of standard normal CDF, numerically stable across the whole range.
__device__ __forceinline__ float log_ndtr(float t) {
    const float IS2 = 0.70710678118654752f;
    if (t > 5.0f) {
        // log(1 - Q) ~= -Q, Q = 0.5*erfc(t/sqrt2) tiny
        return -0.5f * erfcf(t * IS2);
    }
    if (t > -12.0f) {
        // direct: erfc is well scaled here (no underflow until ~ -13)
        return __logf(0.5f * erfcf(-t * IS2));
    }
    // asymptotic: Phi(t) = phi(t)/(-t) * (1 - 1/t^2 + 3/t^4 - 15/t^6 + 105/t^8)
    float w   = 1.0f / (t * t);                 // t*t may overflow -> w=0, fine
    float ser = 1.0f + w * (-1.0f + w * (3.0f + w * (-15.0f + w * 105.0f)));
    return fmaf(-0.5f * t, t, -LNS2PI) - __logf(-t) + __logf(ser);
}

// ln Pr(X \cap Y) for one pair: 256-sample log-domain integral.
__device__ float ln_intersect_pair(float xi, float yj, float cosv) {
    const float delta = MAXV / (float)(NSAMP - 1);
    float s2   = fminf(fmaxf(1.0f - cosv * cosv, 0.0f), 1.0f);
    float sinv = sqrtf(s2);
    float csc  = fminf(1.0f / sinv, INFC);      // 1/0 = +inf -> clamp to 1e10
    float x_r  = fmaxf(xi, yj);
    float y_r  = fminf(xi, yj);
    float a    = lnpdf(x_r);
    float b    = log_ndtr(-(y_r - x_r * cosv) * csc);
    float ab   = a + b;
    float acc  = 0.0f;
#pragma unroll 4
    for (int s = 0; s < NSAMP; ++s) {
        float p = (float)s * delta;
        float g = (y_r - p * cosv) * csc;
        float d = log_ndtr(-g);
        float e = __expf(lnpdf(p) + d - ab);
        acc += (p > x_r) ? e : 0.0f;
    }
    return ab + __logf(acc) + LNDELTA;
}

// ---------------------------------------------------------------------------
// Phase 1: row norms for x_vec and y_vec. One wave32 per row.
// ---------------------------------------------------------------------------
__global__ void norms_kernel(const float* __restrict__ xv, const float* __restrict__ yv,
                             float* __restrict__ xn, float* __restrict__ yn,
                             float* __restrict__ rxn, float* __restrict__ ryn) {
    int wave = (blockIdx.x * blockDim.x + threadIdx.x) >> 5;   // 0..1023
    int lane = threadIdx.x & 31;
    const float* src = (wave < NX) ? (xv + wave * KDIM) : (yv + (wave - NX) * KDIM);
    float a0 = src[lane];
    float a1 = src[lane + 32];
    float s  = fmaf(a0, a0, a1 * a1);
#pragma unroll
    for (int m = 16; m >= 1; m >>= 1) s += __shfl_xor(s, m, 32);
    if (lane == 0) {
        float n = sqrtf(s);
        if (wave < NX) { xn[wave] = n;       rxn[wave] = 1.0f / n; }
        else           { yn[wave - NX] = n;  ryn[wave - NX] = 1.0f / n; }
    }
}

// ---------------------------------------------------------------------------
// Phase 2: min over the 512 x-norms (single block).
// ---------------------------------------------------------------------------
__global__ void min_kernel(const float* __restrict__ xn, float* __restrict__ xmin) {
    __shared__ float sm[256];
    int t = threadIdx.x;
    sm[t] = fminf(xn[t], xn[t + 256]);
    __syncthreads();
    for (int s = 128; s > 0; s >>= 1) {
        if (t < s) sm[t] = fminf(sm[t], sm[t + s]);
        __syncthreads();
    }
    if (t == 0) xmin[0] = sm[0];
}

// ---------------------------------------------------------------------------
// Phase 3: cos = (Xv @ Yv^T) * rnx * rny via V_WMMA_F32_16X16X4_F32.
// One wave per 16x16 tile; 16 K-steps of K=4.
//
// f32 A-fragment (16x4): lanes 0-15 hold M=0..15 at K = {k, k+1},
//                        lanes 16-31 hold M=0..15 at K = {k+2, k+3}.
// B = Yv^T (4x16, column-major per N) has the identical per-lane indexing
// on Yv rows, so both fragments are 8-byte float2 loads off the row base.
// C/D layout: VGPR r -> M = r (lanes 0-15) / r+8 (lanes 16-31), N = lane&15.
// ---------------------------------------------------------------------------
__global__ void cos_wmma_kernel(const float* __restrict__ xv, const float* __restrict__ yv,
                                const float* __restrict__ rxn, const float* __restrict__ ryn,
                                float* __restrict__ cosm) {
    int lane = threadIdx.x;
    int l    = lane & 15;
    int half = lane >> 4;
    int i0   = blockIdx.x * 16;
    int j0   = blockIdx.y * 16;

    const v2f* arow = (const v2f*)(xv + (i0 + l) * KDIM);
    const v2f* brow = (const v2f*)(yv + (j0 + l) * KDIM);

    v8f acc = {0.f, 0.f, 0.f, 0.f, 0.f, 0.f, 0.f, 0.f};
#pragma unroll
    for (int k2 = 0; k2 < KDIM / 2; k2 += 2) {      // k2 = k/2, K-step of 4
        v2f af = arow[k2 + half];
        v2f bf = brow[k2 + half];
        acc = __builtin_amdgcn_wmma_f32_16x16x4_f32(
            /*neg_a=*/false, af, /*neg_b=*/false, bf,
            /*c_mod=*/(short)0, acc, /*reuse_a=*/false, /*reuse_b=*/false);
    }

    int   n  = j0 + l;
    float rn = ryn[n];
#pragma unroll
    for (int r = 0; r < 8; ++r) {
        int   m = i0 + r + half * 8;
        float c = acc[r] * rxn[m] * rn;
        c = fminf(fmaxf(c, -1.0f), 1.0f);
        if (c > 1.0f - 1e-6f) c = 1.0f;   // exact-equal rows -> cos = 1
        cosm[m * NY + n] = c;
    }
}

// ---------------------------------------------------------------------------
// Phase 4: ln_x[i] = ln_intersection(x, xmin/2, cos=1)[:,0]
// cos=1 -> sin=0 -> csc clamps to 1e10, matching the reference exactly.
// ---------------------------------------------------------------------------
__global__ void lnx_kernel(const float* __restrict__ xn, const float* __restrict__ xmin,
                           float* __restrict__ lnx) {
    int i = blockIdx.x * blockDim.x + threadIdx.x;
    if (i >= NX) return;
    float yc = xmin[0] * 0.5f;
    lnx[i] = ln_intersect_pair(xn[i], yc, 1.0f);
}

// ---------------------------------------------------------------------------
// Phase 5: one thread per (i,j) pair; transcendental-bound hot loop.
// ---------------------------------------------------------------------------
__global__ void pair_kernel(const float* __restrict__ cosm, const float* __restrict__ xn,
                            const float* __restrict__ yn, const float* __restrict__ lnx,
                            float* __restrict__ out) {
    int idx = blockIdx.x * blockDim.x + threadIdx.x;
    int i = idx >> 9;          // /512
    int j = idx & (NY - 1);
    float li = ln_intersect_pair(xn[i], yn[j], cosm[idx]);
    float lp = li - lnx[i];
    lp = fminf(fmaxf(lp, -1.0e4f), -1.0e-7f);
    out[idx] = lp - __logf(-expm1f(lp));
}

// ---------------------------------------------------------------------------
// launch
// ---------------------------------------------------------------------------
extern "C" void kernel_launch(void* const* d_in, const int* in_sizes, int n_in,
                              void* d_out, int out_size, void* d_ws, size_t ws_size,
                              hipStream_t stream) {
    const float* xv = (const float*)d_in[0];
    const float* yv = (const float*)d_in[1];

    float* ws   = (float*)d_ws;
    float* cosm = ws;                       // 512*512        = 262144 floats
    float* xn   = ws + (size_t)NX * NY;     // 512
    float* yn   = xn + NX;                  // 512
    float* rxn  = yn + NY;                  // 512
    float* ryn  = rxn + NX;                 // 512
    float* lnx  = ryn + NY;                 // 512
    float* xmin = lnx + NX;                 // 1

    norms_kernel<<<128, 256, 0, stream>>>(xv, yv, xn, yn, rxn, ryn);
    min_kernel<<<1, 256, 0, stream>>>(xn, xmin);
    cos_wmma_kernel<<<dim3(NX / 16, NY / 16), 32, 0, stream>>>(xv, yv, rxn, ryn, cosm);
    lnx_kernel<<<2, 256, 0, stream>>>(xn, xmin, lnx);
    pair_kernel<<<(NX * NY) / 256, 256, 0, stream>>>(cosm, xn, yn, lnx, (float*)d_out);
}